// GCN_25709674234530
// MI455X (gfx1250) — compile-verified
//
#include <hip/hip_runtime.h>
#include <math.h>

typedef __attribute__((ext_vector_type(2))) float v2f;
typedef __attribute__((ext_vector_type(8))) float v8f;

#define N_NODES 500000
#define KNN     8
#define NEDGE   (N_NODES * KNN)      // 4,000,000
#define F_IN    8
#define H1      64
#define H2      16
#define NTILES  (N_NODES / 16)       // 31250 exactly, no tail

// ---------------- degree / normalization ----------------

__global__ void gcn_init_deg(float* __restrict__ deg) {
    int i = blockIdx.x * 256 + threadIdx.x;
    if (i < N_NODES) deg[i] = 1.0f;                 // self-loop
}

__global__ void gcn_accum_deg(const int* __restrict__ col, float* __restrict__ deg) {
    int e = blockIdx.x * 256 + threadIdx.x;
    if (e < NEDGE) atomicAdd(&deg[col[e]], 1.0f);
}

__global__ void gcn_rsqrt_deg(float* __restrict__ deg) {
    int i = blockIdx.x * 256 + threadIdx.x;
    if (i < N_NODES) deg[i] = rsqrtf(deg[i]);       // deg >= 1 always
}

// ---------------- layer 1 pre-scale: xs = dinv[n] * x[n]  (N x 8) ----------------

__global__ void gcn_prescale_x(const float* __restrict__ x, const float* __restrict__ dinv,
                               float* __restrict__ xs) {
    int tid = blockIdx.x * 256 + threadIdx.x;       // N*8 threads
    if (tid >= N_NODES * F_IN) return;
    xs[tid] = dinv[tid >> 3] * x[tid];
}

// ---------------- layer 1 scatter in INPUT space (8 floats/edge, 32M atomics) ----------------
// aggregation commutes with the linear transform: aggregate x first, transform after.

__global__ void gcn_scatter_x(const int* __restrict__ rowIdx, const int* __restrict__ colIdx,
                              const float* __restrict__ xs, float* __restrict__ aggx) {
    int tid = blockIdx.x * 256 + threadIdx.x;       // NEDGE*2 threads (two float4 chunks)
    if (tid >= NEDGE * 2) return;
    const int e  = tid >> 1;
    const int cc = tid & 1;
    const int r = rowIdx[e], c = colIdx[e];
    const float4 v = *(const float4*)(xs + (size_t)r * F_IN + cc * 4);
    float* dst = aggx + (size_t)c * F_IN + cc * 4;  // 16MB region: L2-resident atomics
    atomicAdd(dst + 0, v.x);
    atomicAdd(dst + 1, v.y);
    atomicAdd(dst + 2, v.z);
    atomicAdd(dst + 3, v.w);
}

// ---------------- layer 1 fused WMMA: h1 = relu( [dinv*(aggx+xs)] @ W1 + b1 ) ----------------
// A = z (16x8 f32 tile, built on the fly), B = W1 16-col chunk, 2 chained 16x16x4 WMMAs.
// block = 128 (4 waves); wave w computes cols [16w,16w+16).

__global__ void gcn_layer1_wmma(const float* __restrict__ aggx, const float* __restrict__ xs,
                                const float* __restrict__ W1, const float* __restrict__ b1,
                                const float* __restrict__ dinv, float* __restrict__ h1) {
    const int lane = threadIdx.x & 31;
    const int wave = threadIdx.x >> 5;      // 0..3 -> 16-col chunk of H1=64
    const int tile = blockIdx.x;
    const int m  = lane & 15;
    const int hi = lane >> 4;               // 0: K={0,1}, 1: K={2,3}
    const int node = tile * 16 + m;
    const int col  = wave * 16 + m;         // n = lane&15 for B and D
    const float dv = dinv[node];

    v8f c = {};
#pragma unroll
    for (int kb = 0; kb < F_IN; kb += 4) {
        const int k0 = kb + hi * 2;
        v2f a;                              // A row m: z = dinv*(aggx + xs)
        a.x = dv * (aggx[node * F_IN + k0]     + xs[node * F_IN + k0]);
        a.y = dv * (aggx[node * F_IN + k0 + 1] + xs[node * F_IN + k0 + 1]);
        v2f b;                              // B col n
        b.x = W1[(k0)     * H1 + col];
        b.y = W1[(k0 + 1) * H1 + col];
        c = __builtin_amdgcn_wmma_f32_16x16x4_f32(false, a, false, b, (short)0, c,
                                                  false, false);
    }
    const float bias = b1[col];
#pragma unroll
    for (int r = 0; r < 8; ++r) {           // D: VGPR r -> row r (+8 for hi lanes)
        const int rowNode = tile * 16 + r + hi * 8;
        h1[rowNode * H1 + wave * 16 + m] = fmaxf(c[r] + bias, 0.0f);
    }
}

// ---------------- layer 2 dense: g2 = dinv * (h1 @ W2), K=64 chained WMMA ----------------
// H2=16 < H1=64, so transform-first minimizes scatter width here.

__global__ void gcn_hw2_wmma(const float* __restrict__ h1, const float* __restrict__ W2,
                             const float* __restrict__ dinv, float* __restrict__ g2) {
    const int lane = threadIdx.x & 31;
    const int tile = blockIdx.x;
    const int m  = lane & 15;
    const int hi = lane >> 4;
    const int node = tile * 16 + m;

    v8f c = {};
#pragma unroll
    for (int kc = 0; kc < H1 / 4; ++kc) {
        const int k0 = kc * 4 + hi * 2;
        v2f a;
        a.x = h1[node * H1 + k0];
        a.y = h1[node * H1 + k0 + 1];
        v2f b;
        b.x = W2[(k0)     * H2 + m];
        b.y = W2[(k0 + 1) * H2 + m];
        c = __builtin_amdgcn_wmma_f32_16x16x4_f32(false, a, false, b, (short)0, c,
                                                  false, false);
    }
#pragma unroll
    for (int r = 0; r < 8; ++r) {
        const int rowNode = tile * 16 + r + hi * 8;
        g2[rowNode * H2 + m] = dinv[rowNode] * c[r];
    }
}

// ---------------- layer 2 scatter: agg2[col] += g2[row], 16 floats/edge ----------------

__global__ void gcn_scatter2(const int* __restrict__ rowIdx, const int* __restrict__ colIdx,
                             const float* __restrict__ g2, float* __restrict__ agg2) {
    int tid = blockIdx.x * 256 + threadIdx.x;       // NEDGE*4 threads
    if (tid >= NEDGE * 4) return;
    const int e  = tid >> 2;
    const int cc = tid & 3;
    const int r = rowIdx[e], c = colIdx[e];
    const float4 v = *(const float4*)(g2 + (size_t)r * H2 + cc * 4);
    float* dst = agg2 + (size_t)c * H2 + cc * 4;    // 32MB region: L2-resident atomics
    atomicAdd(dst + 0, v.x);
    atomicAdd(dst + 1, v.y);
    atomicAdd(dst + 2, v.z);
    atomicAdd(dst + 3, v.w);
}

// ---------------- fused tail: h2 = relu(dinv*(agg2+g2)+b2); out = sigmoid(h2.Wl + bl) ----------------

__global__ void gcn_tail(const float* __restrict__ agg2, const float* __restrict__ g2,
                         const float* __restrict__ b2, const float* __restrict__ dinv,
                         const float* __restrict__ Wl, const float* __restrict__ bl,
                         float* __restrict__ out) {
    int n = blockIdx.x * 256 + threadIdx.x;
    if (n >= N_NODES) return;
    const float dv = dinv[n];
    const float4* ap = (const float4*)(agg2 + (size_t)n * H2);
    const float4* gp = (const float4*)(g2   + (size_t)n * H2);
    const float4* bp = (const float4*)b2;
    const float4* wp = (const float4*)Wl;
    float z = bl[0];
#pragma unroll
    for (int q = 0; q < 4; ++q) {
        float4 a = ap[q], g = gp[q], b = bp[q], w = wp[q];
        z += fmaxf(dv * (a.x + g.x) + b.x, 0.0f) * w.x;
        z += fmaxf(dv * (a.y + g.y) + b.y, 0.0f) * w.y;
        z += fmaxf(dv * (a.z + g.z) + b.z, 0.0f) * w.z;
        z += fmaxf(dv * (a.w + g.w) + b.w, 0.0f) * w.w;
    }
    out[n] = 1.0f / (1.0f + expf(-z));
}

// ---------------- launcher ----------------

extern "C" void kernel_launch(void* const* d_in, const int* in_sizes, int n_in,
                              void* d_out, int out_size, void* d_ws, size_t ws_size,
                              hipStream_t stream) {
    const float* x    = (const float*)d_in[0];
    const int*   ei   = (const int*)d_in[1];        // [2, E]
    const float* W1   = (const float*)d_in[2];
    const float* b1   = (const float*)d_in[3];
    const float* W2   = (const float*)d_in[4];
    const float* b2   = (const float*)d_in[5];
    const float* Wl   = (const float*)d_in[6];
    const float* bl   = (const float*)d_in[7];
    float*       out  = (float*)d_out;

    const int* rowIdx = ei;             // sources
    const int* colIdx = ei + NEDGE;     // targets

    // workspace layout (floats):
    //   dinv[N] | bufA[N*16] (xs = bufA[0..N*8), aggx = bufA[N*8..N*16)) | h1[N*64] | agg2[N*16]
    // after layer-1 WMMA, bufA is dead -> g2 (N*16) aliases bufA.
    float* dinv = (float*)d_ws;
    float* bufA = dinv + N_NODES;
    float* xs   = bufA;
    float* aggx = bufA + (size_t)N_NODES * F_IN;
    float* h1   = bufA + (size_t)N_NODES * H2;
    float* agg2 = h1 + (size_t)N_NODES * H1;
    float* g2   = bufA;                 // reuse

    const int B = 256;

    // normalization
    gcn_init_deg <<<(N_NODES + B - 1) / B, B, 0, stream>>>(dinv);
    gcn_accum_deg<<<(NEDGE   + B - 1) / B, B, 0, stream>>>(colIdx, dinv);
    gcn_rsqrt_deg<<<(N_NODES + B - 1) / B, B, 0, stream>>>(dinv);

    // layer 1: pre-scale, aggregate in input space (8-wide), fused WMMA transform + bias + relu
    hipMemsetAsync(aggx, 0, (size_t)N_NODES * F_IN * sizeof(float), stream);
    gcn_prescale_x<<<(N_NODES * F_IN + B - 1) / B, B, 0, stream>>>(x, dinv, xs);
    gcn_scatter_x <<<(NEDGE * 2 + B - 1) / B, B, 0, stream>>>(rowIdx, colIdx, xs, aggx);
    gcn_layer1_wmma<<<NTILES, 128, 0, stream>>>(aggx, xs, W1, b1, dinv, h1);

    // layer 2: transform-first (16-wide messages), scatter, fused finalize + head
    hipMemsetAsync(agg2, 0, (size_t)N_NODES * H2 * sizeof(float), stream);
    gcn_hw2_wmma<<<NTILES, 32, 0, stream>>>(h1, W2, dinv, g2);   // g2 overwrites dead xs/aggx
    gcn_scatter2<<<(NEDGE * 4 + B - 1) / B, B, 0, stream>>>(rowIdx, colIdx, g2, agg2);
    gcn_tail    <<<(N_NODES + B - 1) / B, B, 0, stream>>>(agg2, g2, b2, dinv, Wl, bl, out);
}